// DeepSeekGate_85624468013210
// MI455X (gfx1250) — compile-verified
//
#include <hip/hip_runtime.h>
#include <math.h>
#include <stdint.h>

typedef __attribute__((ext_vector_type(2))) float v2f;
typedef __attribute__((ext_vector_type(4))) float v4f;
typedef __attribute__((ext_vector_type(8))) float v8f;
typedef __attribute__((ext_vector_type(4))) unsigned int u32x4;
typedef __attribute__((ext_vector_type(4))) int i32x4;
typedef __attribute__((ext_vector_type(8))) int i32x8;

#define TPB_TOKENS 16       // tokens per block (WMMA M)
#define DIM        2048     // K
#define NEXP       64       // experts: 4 wave-blocks of 16 (WMMA N)
#define KCHUNK     256      // K per staged chunk
#define NCHUNK     (DIM / KCHUNK)
#define LOG_STRIDE 65       // padded logits stride
#define TOPK       6
#define NSHARED    2
#define NTOK       (4 * 4096)
#define RSCALE     2.5f

#if defined(__has_builtin)
#if __has_builtin(__builtin_amdgcn_tensor_load_to_lds)
#define USE_TDM 1
#endif
#endif
#ifndef USE_TDM
#define USE_TDM 0
#endif

#if USE_TDM
// TDM pads +4 dwords every 256 dwords stored -> LDS row stride 260 floats.
// 260 % 64banks = 4 -> lane r reads banks {4r..4r+3}: conflict-free b64 A-frags.
#define XSTRIDE 260
#else
#define XSTRIDE 264         // 264*4=1056B rows: 16B-aligned b128 stores, 2-way conflict
#endif

#if defined(__has_builtin) && __has_builtin(__builtin_amdgcn_s_wait_tensorcnt)
#define WAIT_TENSORCNT0() __builtin_amdgcn_s_wait_tensorcnt(0)
#else
#define WAIT_TENSORCNT0() asm volatile("s_wait_tensorcnt 0x0" ::: "memory")
#endif

#if USE_TDM
__device__ __forceinline__ uint32_t lds_byte_off(const void* p) {
    // generic -> LDS address space; ptr value is byte offset within LDS alloc
    return (uint32_t)(uintptr_t)(const __attribute__((address_space(3))) void*)p;
}

// DMA a [16 x KCHUNK] f32 tile (row pitch DIM) from global into LDS.
__device__ __forceinline__ void tdm_load_tile(uint32_t lds_byte, unsigned long long gaddr) {
    u32x4 g0;
    g0[0] = 1u;                                       // count=1, user descriptor
    g0[1] = lds_byte;                                 // lds_addr [63:32]
    g0[2] = (uint32_t)gaddr;                          // global_addr lo
    g0[3] = (uint32_t)((gaddr >> 32) & 0x01FFFFFFull) // global_addr [56:32]
          | (2u << 30);                               // type=2 (image)
    i32x8 g1;
    g1[0] = (2 << 16)       // data_size = 4B
          | (1 << 20)       // pad_enable
          | (7 << 22)       // pad_interval code 7 = 256 dwords
          | (3 << 25);      // pad_amount  code 3 = 4 dwords
    g1[1] = (int)((unsigned)DIM << 16);        // tensor_dim0 lo16 in [63:48]
    g1[2] = (int)((unsigned)NTOK << 16);       // tensor_dim1 lo16 in [95:80]
    g1[3] = (int)((unsigned)KCHUNK << 16);     // tile_dim0 in [127:112]
    g1[4] = TPB_TOKENS;                        // tile_dim1 in [143:128]
    g1[5] = DIM;                               // tensor_dim0_stride lo32
    g1[6] = 0;
    g1[7] = 0;
    i32x4 z4 = {0, 0, 0, 0};
#if __clang_major__ >= 23
    i32x8 z8 = {0, 0, 0, 0, 0, 0, 0, 0};
    __builtin_amdgcn_tensor_load_to_lds(g0, g1, z4, z4, z8, 0);
#else
    __builtin_amdgcn_tensor_load_to_lds(g0, g1, z4, z4, 0);
#endif
}
#endif // USE_TDM

__global__ __launch_bounds__(128) void deepseek_gate_kernel(
    const float* __restrict__ x,
    const float* __restrict__ W,
    const float* __restrict__ gate_bias,
    float* __restrict__ out_w,
    int* __restrict__ out_idx)
{
    __shared__ float sX[2][TPB_TOKENS * XSTRIDE];    // double-buffered x tile
    __shared__ float sLog[TPB_TOKENS * LOG_STRIDE];  // logits -> scores
    __shared__ float sBias[TPB_TOKENS * LOG_STRIDE]; // biased scores (top-k scratch)

    const int tid  = threadIdx.x;
    const int wave = tid >> 5;     // 0..3 -> expert block
    const int lane = tid & 31;     // wave32
    const int tb   = blockIdx.x * TPB_TOKENS;
    const int e_base = wave * 16;

    // V_WMMA_F32_16X16X4_F32 fragment lane mapping:
    //  A 16x4: lanes 0-15 -> M=lane, K0/K1 ; lanes 16-31 -> M=lane-16, K2/K3
    //  B 4x16: lanes 0-15 -> N=lane, K0/K1 ; lanes 16-31 -> N=lane-16, K2/K3
    const int a_row  = lane & 15;
    const int a_koff = (lane >> 4) * 2;
    const float* wrow = W + (size_t)(e_base + (lane & 15)) * DIM + a_koff;

    v8f acc = {};

#if USE_TDM
    const unsigned long long xbase =
        (unsigned long long)(uintptr_t)(x + (size_t)tb * DIM);
    if (wave == 0) {
        tdm_load_tile(lds_byte_off(&sX[0][0]), xbase);
    }
#endif

    for (int i = 0; i < NCHUNK; ++i) {
        float* cur = &sX[i & 1][0];

#if USE_TDM
        if (wave == 0) {
            WAIT_TENSORCNT0();                 // chunk i landed in LDS
        }
        __syncthreads();                       // publish to all waves; also fences
                                               // everyone past compute(i-1)
        if (wave == 0 && (i + 1) < NCHUNK) {   // DMA chunk i+1 under compute i
            tdm_load_tile(lds_byte_off(&sX[(i + 1) & 1][0]),
                          xbase + (unsigned long long)(i + 1) * KCHUNK * 4ull);
        }
#else
        __syncthreads();                       // protect prior reads of this buffer
        #pragma unroll
        for (int j = 0; j < 8; ++j) {          // 16 rows x 256 f32 = 1024 float4
            int f4  = tid + j * 128;
            int row = f4 >> 6;                 // 64 float4 per row
            int c4  = f4 & 63;
            v4f v = *(const v4f*)(x + (size_t)(tb + row) * DIM + i * KCHUNK + c4 * 4);
            *(v4f*)(&cur[row * XSTRIDE + c4 * 4]) = v;
        }
        __syncthreads();
#endif

        // 4 groups of K=64: batch-load 16 A-frags + 16 B-frags, then 16 dense WMMAs
        #pragma unroll
        for (int kg = 0; kg < KCHUNK / 64; ++kg) {
            const float* abase = cur + a_row * XSTRIDE + kg * 64 + a_koff;
            const float* bbase = wrow + i * KCHUNK + kg * 64;
            v2f afrag[16], bfrag[16];
            #pragma unroll
            for (int j = 0; j < 16; ++j) afrag[j] = *(const v2f*)(abase + 4 * j);
            #pragma unroll
            for (int j = 0; j < 16; ++j) bfrag[j] = *(const v2f*)(bbase + 4 * j);
            #pragma unroll
            for (int j = 0; j < 16; ++j)
                acc = __builtin_amdgcn_wmma_f32_16x16x4_f32(
                    false, afrag[j], false, bfrag[j], (short)0, acc, false, false);
        }
    }

    // scatter C (16x16 f32): VGPR r, lane l -> M = r + 8*(l>=16), N = l&15
    {
        const int m_hi = (lane >> 4) * 8;
        const int n    = e_base + (lane & 15);
        #pragma unroll
        for (int r = 0; r < 8; ++r)
            sLog[(r + m_hi) * LOG_STRIDE + n] = acc[r];
    }
    __syncthreads();

    // epilogue: one thread per token (64 sigmoids + top-6 select + normalize)
    if (tid < TPB_TOKENS) {
        float* srow = &sLog[tid * LOG_STRIDE];   // overwrite logits with scores
        float* brow = &sBias[tid * LOG_STRIDE];
        #pragma unroll 8
        for (int e = 0; e < NEXP; ++e) {
            float s = 1.0f / (1.0f + expf(-srow[e]));   // sigmoid score
            srow[e] = s;
            brow[e] = s + gate_bias[e];                 // bias affects selection only
        }

        int   sel[TOPK];
        float wsel[TOPK];
        float wsum = 0.0f;
        #pragma unroll
        for (int k = 0; k < TOPK; ++k) {
            int best = 0;
            float bv = brow[0];
            for (int e = 1; e < NEXP; ++e) {
                float v = brow[e];
                if (v > bv) { bv = v; best = e; }  // strict > == lowest-index tie-break
            }
            sel[k]  = best;
            wsel[k] = srow[best];
            wsum   += srow[best];
            brow[best] = -INFINITY;
        }

        const float inv = RSCALE / wsum;
        const size_t tk = (size_t)tb + tid;
        float* wp = out_w + tk * TOPK;
        #pragma unroll
        for (int k = 0; k < TOPK; ++k) wp[k] = wsel[k] * inv;

        int* ip = out_idx + tk * (TOPK + NSHARED);
        ip[0] = 0;
        ip[1] = 1;
        #pragma unroll
        for (int k = 0; k < TOPK; ++k) ip[2 + k] = sel[k] + NSHARED;
    }
}

extern "C" void kernel_launch(void* const* d_in, const int* in_sizes, int n_in,
                              void* d_out, int out_size, void* d_ws, size_t ws_size,
                              hipStream_t stream) {
    const float* x  = (const float*)d_in[0];   // [4,4096,2048] f32
    const float* W  = (const float*)d_in[1];   // [64,2048] f32
    const float* gb = (const float*)d_in[2];   // [64] f32

    float* out_w   = (float*)d_out;                               // [NTOK,6] f32
    int*   out_idx = (int*)((float*)d_out + (size_t)NTOK * TOPK); // [NTOK,8] i32

    deepseek_gate_kernel<<<NTOK / TPB_TOKENS, 128, 0, stream>>>(x, W, gb, out_w, out_idx);
}